// ManifoldGPRFilter_38062000177650
// MI455X (gfx1250) — compile-verified
//
#include <hip/hip_runtime.h>
#include <math.h>

#define D 64
#define EPSF 1e-8f
#define TILE_ROWS 128
#define ROW_STRIDE 68   // padded LDS row stride (floats): bank-conflict-free A reads

typedef __attribute__((ext_vector_type(2))) float v2f;
typedef __attribute__((ext_vector_type(8))) float v8f;

// ---------------- graph preprocessing (once per launch) ----------------

__global__ void zero_counts_kernel(int* __restrict__ counts, int n) {
  int i = blockIdx.x * blockDim.x + threadIdx.x;
  if (i < n) counts[i] = 0;
}

__global__ void count_kernel(const int* __restrict__ dst, int* __restrict__ counts, int E) {
  int e = blockIdx.x * blockDim.x + threadIdx.x;
  if (e < E) atomicAdd(&counts[dst[e]], 1);
}

// single-workgroup exclusive scan (n ~ 1e5, runs once; carries across 1024-chunks)
__global__ void __launch_bounds__(1024) scan_kernel(const int* __restrict__ counts,
                                                    int* __restrict__ row_start,
                                                    int* __restrict__ cursor, int n) {
  __shared__ int sdata[1024];
  __shared__ int carry_s;
  const int tid = threadIdx.x;
  if (tid == 0) carry_s = 0;
  __syncthreads();
  for (int base = 0; base < n; base += 1024) {
    const int i = base + tid;
    const int v = (i < n) ? counts[i] : 0;
    sdata[tid] = v;
    __syncthreads();
    for (int off = 1; off < 1024; off <<= 1) {
      int t = (tid >= off) ? sdata[tid - off] : 0;
      __syncthreads();
      sdata[tid] += t;
      __syncthreads();
    }
    const int incl = sdata[tid];
    const int carry = carry_s;
    if (i < n) {
      const int excl = carry + incl - v;
      row_start[i] = excl;
      cursor[i]    = excl;
    }
    __syncthreads();                    // all lanes have read carry_s
    if (tid == 1023) carry_s += sdata[1023];
    __syncthreads();
  }
  if (tid == 0) row_start[n] = carry_s;
}

__global__ void scatter_kernel(const int* __restrict__ src, const int* __restrict__ dst,
                               const float* __restrict__ val, int* __restrict__ cursor,
                               int* __restrict__ csc_src, float* __restrict__ csc_val, int E) {
  int e = blockIdx.x * blockDim.x + threadIdx.x;
  if (e >= E) return;
  int d = dst[e];
  int pos = atomicAdd(&cursor[d], 1);
  csc_src[pos] = src[e];
  csc_val[pos] = val[e];
}

// ---------------- init: Hcur = X, Z = w0*X ----------------

__global__ void init_kernel(const float* __restrict__ X, const float* __restrict__ w,
                            float* __restrict__ Z, float* __restrict__ Hcur, int total) {
  int i = blockIdx.x * blockDim.x + threadIdx.x;
  float w0 = w[0];
  if (i < total) {
    float x = X[i];
    Hcur[i] = x;
    Z[i] = w0 * x;
  }
}

// ---------------- SpMM: one wave per dst row, 2 features per lane ----------------

__global__ void spmm_gather_kernel(const float* __restrict__ Hcur, float* __restrict__ Hraw,
                                   const int* __restrict__ row_start,
                                   const int* __restrict__ csc_src,
                                   const float* __restrict__ csc_val, int n) {
  int r    = (blockIdx.x * blockDim.x + threadIdx.x) >> 5;   // wave32
  int lane = threadIdx.x & 31;
  if (r >= n) return;
  int beg = row_start[r], end = row_start[r + 1];
  float ax = 0.f, ay = 0.f;
  for (int j = beg; j < end; ++j) {
    int   s = csc_src[j];
    float v = csc_val[j];
    float2 h = *(const float2*)(Hcur + (size_t)s * D + lane * 2);  // coalesced 256B/wave
    ax = fmaf(v, h.x, ax);
    ay = fmaf(v, h.y, ay);
  }
  float2 o; o.x = ax; o.y = ay;
  *(float2*)(Hraw + (size_t)r * D + lane * 2) = o;
}

// ---------------- norm + accumulate: WMMA Gram-matrix row norms ----------------
// Each wave owns 16 rows; Gram = H_tile * H_tile^T via v_wmma_f32_16x16x4_f32.
// A(16x4) layout: lane l -> M=l%16, VGPR v -> K=2*(l/16)+v.
// B(4x16) layout: lane l -> N=l%16, VGPR v -> K=v+2*(l/16)  (transpose of A) ->
// for B = A^T the per-lane register contents are IDENTICAL, so pass `a` twice.

__global__ void __launch_bounds__(256) norm_kernel(const float* __restrict__ Hraw,
                                                   float* __restrict__ Hcur,
                                                   float* __restrict__ Z,
                                                   const float* __restrict__ w,
                                                   int l, int n_rows) {
  __shared__ float tile[TILE_ROWS * ROW_STRIDE];
  __shared__ float norm2[TILE_ROWS];

  const int row0 = blockIdx.x * TILE_ROWS;
  const int tid  = threadIdx.x;

  // stage 128x64 tile into LDS (coalesced), zero-pad rows beyond N
  for (int i = tid; i < TILE_ROWS * D; i += 256) {
    int r = i >> 6, cidx = i & 63;
    int gr = row0 + r;
    tile[r * ROW_STRIDE + cidx] = (gr < n_rows) ? Hraw[(size_t)gr * D + cidx] : 0.0f;
  }
  __syncthreads();

  const int wave = tid >> 5;
  const int lane = tid & 31;
  const int wrow = wave * 16;
  const int m = lane & 15;
  const int g = lane >> 4;

  v8f c = {};
  const float* base = &tile[(wrow + m) * ROW_STRIDE + 2 * g];
  #pragma unroll
  for (int k = 0; k < D; k += 4) {
    v2f a;
    a.x = base[k];          // A[m][k+2g]   (bank-conflict-free, stride 68)
    a.y = base[k + 1];      // A[m][k+2g+1]
    c = __builtin_amdgcn_wmma_f32_16x16x4_f32(false, a, false, a,
                                              (short)0, c, false, false);
  }

  // diagonal D[i][i] lives in lane (i<8 ? i : i+16), vgpr i%8
  {
    int i = (lane < 8) ? lane : ((lane >= 24) ? (lane - 16) : -1);
    if (i >= 0) {
      int ii = i & 7;
      float dval = c[0];
      dval = (ii == 1) ? c[1] : dval;
      dval = (ii == 2) ? c[2] : dval;
      dval = (ii == 3) ? c[3] : dval;
      dval = (ii == 4) ? c[4] : dval;
      dval = (ii == 5) ? c[5] : dval;
      dval = (ii == 6) ? c[6] : dval;
      dval = (ii == 7) ? c[7] : dval;
      norm2[wrow + i] = dval;
    }
  }
  __syncthreads();

  const float wl = w[l];
  for (int i = tid; i < TILE_ROWS * D; i += 256) {
    int r = i >> 6, cidx = i & 63;
    int gr = row0 + r;
    if (gr < n_rows) {
      float nn = sqrtf(norm2[r]) + EPSF;
      float s  = tanhf(nn) / nn;
      float h  = tile[r * ROW_STRIDE + cidx] * s;
      size_t idx = (size_t)gr * D + cidx;
      Hcur[idx] = h;            // normalized H for next SpMM
      Z[idx]   += wl * h;       // GPR accumulate
    }
  }
}

// ---------------- host-side orchestration ----------------

extern "C" void kernel_launch(void* const* d_in, const int* in_sizes, int n_in,
                              void* d_out, int out_size, void* d_ws, size_t ws_size,
                              hipStream_t stream) {
  const float* X    = (const float*)d_in[0];
  const float* evals= (const float*)d_in[1];
  const float* mw   = (const float*)d_in[2];
  const int*   esrc = (const int*)d_in[3];
  const int*   edst = (const int*)d_in[4];

  const int nD = in_sizes[0];
  const int n  = nD / D;
  const int E  = in_sizes[1];
  const int L  = in_sizes[2] - 1;

  float* Z = (float*)d_out;

  // workspace layout
  float* Hcur      = (float*)d_ws;
  float* Hraw      = Hcur + (size_t)nD;
  int*   counts    = (int*)(Hraw + (size_t)nD);
  int*   row_start = counts + n;            // n+1
  int*   cursor    = row_start + (n + 1);
  int*   csc_src   = cursor + n;
  float* csc_val   = (float*)(csc_src + E);

  // build CSC (by-dst) once; amortized over L SpMM iterations, removes hot-loop atomics
  zero_counts_kernel<<<(n + 255) / 256, 256, 0, stream>>>(counts, n);
  count_kernel<<<(E + 255) / 256, 256, 0, stream>>>(edst, counts, E);
  scan_kernel<<<1, 1024, 0, stream>>>(counts, row_start, cursor, n);
  scatter_kernel<<<(E + 255) / 256, 256, 0, stream>>>(esrc, edst, evals,
                                                      cursor, csc_src, csc_val, E);

  init_kernel<<<(nD + 255) / 256, 256, 0, stream>>>(X, mw, Z, Hcur, nD);

  const int spmm_blocks = (n * 32 + 255) / 256;        // wave32 per row
  const int norm_blocks = (n + TILE_ROWS - 1) / TILE_ROWS;
  for (int l = 1; l <= L; ++l) {
    spmm_gather_kernel<<<spmm_blocks, 256, 0, stream>>>(Hcur, Hraw, row_start,
                                                        csc_src, csc_val, n);
    norm_kernel<<<norm_blocks, 256, 0, stream>>>(Hraw, Hcur, Z, mw, l, n);
  }
}